// TopKSoft_28080496181695
// MI455X (gfx1250) — compile-verified
//
#include <hip/hip_runtime.h>
#include <math.h>

typedef float v4f __attribute__((ext_vector_type(4)));

#define COLS 4096
#define WAVES_PER_BLOCK 8

// Branchless descending top-3 insert (value-only compares -> pure
// v_cmp + v_cndmask, no SALU mask logic, no divergent branches).
// Scanning in ascending index order keeps the lowest index on exact ties.
__device__ __forceinline__ void ins3(float v, int i,
                                     float& v0, int& i0,
                                     float& v1, int& i1,
                                     float& v2, int& i2) {
    const bool b0 = v > v0;
    const bool b1 = v > v1;
    const bool b2 = v > v2;
    const float nv2 = b1 ? v1 : (b2 ? v : v2);
    const int   ni2 = b1 ? i1 : (b2 ? i : i2);
    const float nv1 = b0 ? v0 : (b1 ? v : v1);
    const int   ni1 = b0 ? i0 : (b1 ? i : i1);
    v0 = b0 ? v : v0;
    i0 = b0 ? i : i0;
    v1 = nv1; i1 = ni1;
    v2 = nv2; i2 = ni2;
}

__global__ __launch_bounds__(256) void
TopKSoft_28080496181695_kernel(const float* __restrict__ scores,
                               float* __restrict__ out,
                               int rows) {
    const int lane = threadIdx.x & 31;
    const int wave = threadIdx.x >> 5;
    const int row  = blockIdx.x * WAVES_PER_BLOCK + wave;
    if (row >= rows) return;

    const float* __restrict__ src = scores + (size_t)row * COLS;
    float*       __restrict__ dst = out    + (size_t)row * COLS;

    // ---- Pass 1: per-lane top-3 over 128 elements (32 x float4, coalesced NT b128)
    float v0 = -INFINITY, v1 = -INFINITY, v2 = -INFINITY;
    int   i0 = 0, i1 = 0, i2 = 0;

#pragma unroll 8
    for (int c = 0; c < COLS / (32 * 4); ++c) {
        const int base = (c * 32 + lane) * 4;
        const v4f d = __builtin_nontemporal_load((const v4f*)(src + base));
#pragma unroll
        for (int e = 0; e < 4; ++e) {
            ins3(d[e], base + e, v0, i0, v1, i1, v2, i2);
        }
    }

    // ---- Wave32 xor-butterfly: every lane ends with the row's top-3.
#pragma unroll
    for (int off = 16; off >= 1; off >>= 1) {
        const float ov0 = __shfl_xor(v0, off, 32);
        const int   oi0 = __shfl_xor(i0, off, 32);
        const float ov1 = __shfl_xor(v1, off, 32);
        const int   oi1 = __shfl_xor(i1, off, 32);
        const float ov2 = __shfl_xor(v2, off, 32);
        const int   oi2 = __shfl_xor(i2, off, 32);
        ins3(ov0, oi0, v0, i0, v1, i1, v2, i2);
        ins3(ov1, oi1, v0, i0, v1, i1, v2, i2);
        ins3(ov2, oi2, v0, i0, v1, i1, v2, i2);
    }

    // ---- Softmax over the 3 survivors. Non-top lanes contribute
    //      exp(-1e9 - max) which underflows to exactly 0.0f in fp32,
    //      matching the reference numerics.
    const float e1 = __expf(v1 - v0);
    const float e2 = __expf(v2 - v0);
    const float inv = 1.0f / (1.0f + e1 + e2);
    const float p0 = inv, p1 = e1 * inv, p2 = e2 * inv;

    // ---- Pass 2: stream pure zeros (no per-element ALU work) with NT b128
    //      stores, then patch the 3 nonzero probabilities.
#pragma unroll 8
    for (int c = 0; c < COLS / (32 * 4); ++c) {
        const int base = (c * 32 + lane) * 4;
        const v4f z = {0.0f, 0.0f, 0.0f, 0.0f};
        __builtin_nontemporal_store(z, (v4f*)(dst + base));
    }

    // Drain the zero stores so the scalar fix-up stores below are ordered
    // after them at every cache level (split-counter model: STOREcnt).
    asm volatile("s_wait_storecnt 0x0" ::: "memory");

    if (lane < 3) {
        const int   idx = (lane == 0) ? i0 : (lane == 1) ? i1 : i2;
        const float pv  = (lane == 0) ? p0 : (lane == 1) ? p1 : p2;
        dst[idx] = pv;
    }
}

extern "C" void kernel_launch(void* const* d_in, const int* in_sizes, int n_in,
                              void* d_out, int out_size, void* d_ws, size_t ws_size,
                              hipStream_t stream) {
    (void)n_in; (void)d_ws; (void)ws_size; (void)out_size;
    const float* scores = (const float*)d_in[0];
    float* out = (float*)d_out;

    const int rows   = in_sizes[0] / COLS;                                // 16384
    const int blocks = (rows + WAVES_PER_BLOCK - 1) / WAVES_PER_BLOCK;    // 2048

    TopKSoft_28080496181695_kernel<<<blocks, 256, 0, stream>>>(scores, out, rows);
}